// DWT_51780125721277
// MI455X (gfx1250) — compile-verified
//
#include <hip/hip_runtime.h>

typedef __attribute__((ext_vector_type(4))) float v4f;
typedef __attribute__((ext_vector_type(2))) float v2f;

// 2x2 Haar DWT, memory-bound streaming kernel.
// Input  : x (32,3,512,512) f32 row-major
// Output : ll|lh|hl|hh concatenated, each (32,3,256,256) f32
//
// Thread t handles one float4 column-group of one output row:
//   f4    = t % 128   -> input cols [4*f4 .. 4*f4+3] -> output cols 2*f4, 2*f4+1
//   orow  = (t/128) % 256
//   plane = t / (128*256)   (n*3 + c, 0..95)
// Loads: 2x b128 (even row + odd row), staged through LDS via the CDNA5
// async engine. Stores: 4x nontemporal b64 (one float2 per subband).
__global__ __launch_bounds__(256) void haar_dwt2d_kernel(
    const float* __restrict__ x, float* __restrict__ out)
{
    __shared__ __align__(16) char smem[256 * 32];   // 32 B staging slot per thread

    const int tid = (int)threadIdx.x;
    const int t   = (int)blockIdx.x * 256 + tid;

    const int f4    = t & 127;
    const int orow  = (t >> 7) & 255;
    const int plane = t >> 15;

    // Input addresses (fit comfortably in 32-bit element indices: max ~25.2M)
    const int row0 = plane * (512 * 512) + (2 * orow) * 512 + (f4 << 2);
    const float* ptop = x + row0;
    const float* pbot = x + row0 + 512;

    // LDS staging slot for this lane
    char* slot = smem + tid * 32;
    unsigned ltop = (unsigned)(size_t)slot;   // flat->LDS offset = addr[31:0]
    unsigned lbot = ltop + 16u;

    // CDNA5 async global->LDS copies (tracked with ASYNCcnt)
    asm volatile(
        "global_load_async_to_lds_b128 %0, %1, off\n\t"
        "global_load_async_to_lds_b128 %2, %3, off"
        :
        : "v"(ltop), "v"(ptop), "v"(lbot), "v"(pbot)
        : "memory");
    asm volatile("s_wait_asynccnt 0x0" ::: "memory");

    // Read staged rows back from LDS (ds_load_b128)
    v4f top = *(const v4f*)(slot);
    v4f bot = *(const v4f*)(slot + 16);

    const float a0 = top.x, b0 = top.y, c0 = bot.x, d0 = bot.y;
    const float a1 = top.z, b1 = top.w, c1 = bot.z, d1 = bot.w;

    const float h = 0.5f;
    v2f ll, lh, hl, hh;
    ll.x = h * ( a0 + b0 + c0 + d0);  ll.y = h * ( a1 + b1 + c1 + d1);
    lh.x = h * (-a0 - b0 + c0 + d0);  lh.y = h * (-a1 - b1 + c1 + d1);
    hl.x = h * (-a0 + b0 - c0 + d0);  hl.y = h * (-a1 + b1 - c1 + d1);
    hh.x = h * ( a0 - b0 - c0 + d0);  hh.y = h * ( a1 - b1 - c1 + d1);

    const int S = 32 * 3 * 256 * 256;                 // 6,291,456 elems / subband
    const int o = plane * (256 * 256) + orow * 256 + (f4 << 1);   // even -> 8B aligned

    // Streaming outputs: written once, never re-read -> nontemporal (TH=NT)
    __builtin_nontemporal_store(ll, (v2f*)(out + o));
    __builtin_nontemporal_store(lh, (v2f*)(out + S + o));
    __builtin_nontemporal_store(hl, (v2f*)(out + 2 * S + o));
    __builtin_nontemporal_store(hh, (v2f*)(out + 3 * S + o));
}

extern "C" void kernel_launch(void* const* d_in, const int* in_sizes, int n_in,
                              void* d_out, int out_size, void* d_ws, size_t ws_size,
                              hipStream_t stream) {
    (void)in_sizes; (void)n_in; (void)out_size; (void)d_ws; (void)ws_size;
    const float* x = (const float*)d_in[0];
    float* out = (float*)d_out;

    // 32*3*256*128 = 3,145,728 work items, 256 threads/block -> 12288 blocks exact
    const int nItems = 32 * 3 * 256 * 128;
    dim3 grid(nItems / 256), block(256);
    hipLaunchKernelGGL(haar_dwt2d_kernel, grid, block, 0, stream, x, out);
}